// GeneEncoder_37503654429370
// MI455X (gfx1250) — compile-verified
//
#include <hip/hip_runtime.h>
#include <hip/hip_bf16.h>
#include <math.h>

// ---------------------------------------------------------------------------
// Problem constants (match reference)
// ---------------------------------------------------------------------------
#define N_SRC1 1210000
#define N_DST1 110000
#define N_SRC2 110000
#define N_DST2 10000
#define E1     1100000
#define E2     100000
#define IN_DIM  128
#define HID_DIM 256
#define OUT_DIM 128

typedef __attribute__((ext_vector_type(2))) float v2f;
typedef __attribute__((ext_vector_type(8))) float v8f;

// ---------------------------------------------------------------------------
// Utility: zero a float buffer (scratch is poisoned 0xAA, must re-zero per call)
// ---------------------------------------------------------------------------
__global__ void zero_f32_kernel(float* __restrict__ p, int n) {
    int i = blockIdx.x * blockDim.x + threadIdx.x;
    if (i < n) p[i] = 0.0f;
}

// ---------------------------------------------------------------------------
// Degree counting: one thread per edge, two hardware f32 atomics
// ---------------------------------------------------------------------------
__global__ void degree_kernel(const int* __restrict__ src, const int* __restrict__ dst,
                              float* __restrict__ deg_src, float* __restrict__ deg_dst,
                              int n_edges) {
    int e = blockIdx.x * blockDim.x + threadIdx.x;
    if (e >= n_edges) return;
    __hip_atomic_fetch_add(&deg_src[src[e]], 1.0f, __ATOMIC_RELAXED, __HIP_MEMORY_SCOPE_AGENT);
    __hip_atomic_fetch_add(&deg_dst[dst[e]], 1.0f, __ATOMIC_RELAXED, __HIP_MEMORY_SCOPE_AGENT);
}

// deg -> rsqrt(max(deg, 1))  in place
__global__ void rnorm_kernel(float* __restrict__ deg, int n) {
    int i = blockIdx.x * blockDim.x + threadIdx.x;
    if (i < n) deg[i] = rsqrtf(fmaxf(deg[i], 1.0f));
}

// ---------------------------------------------------------------------------
// Edge aggregation: one wave (32 lanes) per edge; F/32 floats per lane.
//   agg[dst] += feat[src] * rnorm_src[src]
// float4 vector loads, hardware global_atomic_add_f32 scatter.
// ---------------------------------------------------------------------------
template <int F>
__global__ void aggregate_kernel(const float* __restrict__ feat,
                                 const float* __restrict__ rn_src,
                                 const int* __restrict__ src,
                                 const int* __restrict__ dst,
                                 float* __restrict__ agg,
                                 int n_edges) {
    const int e    = blockIdx.x * (blockDim.x >> 5) + (threadIdx.x >> 5);
    const int lane = threadIdx.x & 31;
    if (e >= n_edges) return;

    const int   s  = src[e];
    const int   d  = dst[e];
    const float rs = rn_src[s];
    const float* __restrict__ fp = feat + (long)s * F;
    float* __restrict__       ap = agg + (long)d * F;

#pragma unroll
    for (int c0 = 0; c0 < F; c0 += 128) {
        const int c = c0 + lane * 4;
        float4 v = *(const float4*)(fp + c);
        __hip_atomic_fetch_add(ap + c + 0, v.x * rs, __ATOMIC_RELAXED, __HIP_MEMORY_SCOPE_AGENT);
        __hip_atomic_fetch_add(ap + c + 1, v.y * rs, __ATOMIC_RELAXED, __HIP_MEMORY_SCOPE_AGENT);
        __hip_atomic_fetch_add(ap + c + 2, v.z * rs, __ATOMIC_RELAXED, __HIP_MEMORY_SCOPE_AGENT);
        __hip_atomic_fetch_add(ap + c + 3, v.w * rs, __ATOMIC_RELAXED, __HIP_MEMORY_SCOPE_AGENT);
    }
}

// ---------------------------------------------------------------------------
// Fused WMMA GEMM + sigmoid gate:
//   hd    = (rowscale[m] * A[M,K]) @ W[K,N] + bias[N]
//   alpha = sigmoid(hd @ attn)            (per row)
//   Out   = hd * alpha;  alpha_out[row] = alpha (optional)
//
// f32 data, V_WMMA_F32_16X16X4_F32, wave32.
//   block = 128 threads = 4 waves; block owns a 16-row M tile and the FULL
//   N extent (4 * TPW * 16 == N), so the row-dot for the gate reduces
//   entirely inside the block: shfl_xor within each 16-lane half, then a
//   4x16-float LDS reduction across waves.
//
// Fragment layouts (ISA 7.12.2):
//   A 16x4 f32: lanes 0-15 K={k0,k0+1}, lanes 16-31 K={k0+2,k0+3}; M=lane&15.
//   B mirrors the K split, N = n0 + (lane&15).
//   C/D: vgpr i -> row = i + 8*(lane>=16), col = lane&15.
// All WMMAs run before any divergent code (EXEC all-1s requirement); the
// M/N tile counts divide exactly so the uniform path holds.
// ---------------------------------------------------------------------------
template <int K_DIM, int TPW>
__global__ __launch_bounds__(128)
void gemm_gate_wmma_kernel(const float* __restrict__ A,
                           const float* __restrict__ rowscale,
                           const float* __restrict__ W,
                           const float* __restrict__ bias,
                           const float* __restrict__ attn,
                           float* __restrict__ Out,
                           float* __restrict__ alpha_out,
                           int N) {
    const int wave = threadIdx.x >> 5;   // 0..3
    const int lane = threadIdx.x & 31;
    const int half = lane >> 4;          // 0: K lo pair, 1: K hi pair
    const int l15  = lane & 15;

    const int row0 = blockIdx.x * 16;
    const int n0   = wave * (TPW * 16);

    const int   am = row0 + l15;         // A row for this lane (both halves)
    const float rs = rowscale[am];
    const float* __restrict__ arow = A + (long)am * K_DIM;

    v8f acc[TPW] = {};

    for (int k0 = 0; k0 < K_DIM; k0 += 4) {
        const int ka = k0 + half * 2;
        v2f a;
        a.x = arow[ka]     * rs;
        a.y = arow[ka + 1] * rs;
        // Load all B fragments first (distinct regs -> load clause, hidden
        // latency), then issue the WMMA group.
        v2f b[TPW];
#pragma unroll
        for (int t = 0; t < TPW; ++t) {
            const int n = n0 + t * 16 + l15;
            b[t].x = W[(long)(ka)     * N + n];
            b[t].y = W[(long)(ka + 1) * N + n];
        }
#pragma unroll
        for (int t = 0; t < TPW; ++t) {
            acc[t] = __builtin_amdgcn_wmma_f32_16x16x4_f32(
                false, a, false, b[t], (short)0, acc[t], false, false);
        }
    }

    // ---- epilogue: bias, fused gate, scale, store ----------------------
    __shared__ float red[4][16];     // per-wave partial row dots
    __shared__ float alphas[16];     // per-row gate

    float attn_v[TPW];
#pragma unroll
    for (int t = 0; t < TPW; ++t) {
        const int   n  = n0 + t * 16 + l15;
        const float bv = bias[n];
        attn_v[t] = attn[n];
#pragma unroll
        for (int i = 0; i < 8; ++i) acc[t][i] += bv;   // acc = hd tile
    }

    // partial dot over this wave's 16*TPW columns, rows r = i + half*8
    float part[8];
#pragma unroll
    for (int i = 0; i < 8; ++i) {
        float p = 0.0f;
#pragma unroll
        for (int t = 0; t < TPW; ++t) p += acc[t][i] * attn_v[t];
        // reduce across the 16 lanes of this half (xor masks stay in-half)
#pragma unroll
        for (int off = 1; off < 16; off <<= 1) p += __shfl_xor(p, off, 32);
        part[i] = p;
    }
    if (l15 == 0) {                  // lane 0 and lane 16 -> disjoint rows
#pragma unroll
        for (int i = 0; i < 8; ++i) red[wave][i + half * 8] = part[i];
    }
    __syncthreads();

    if (threadIdx.x < 16) {
        const int r = threadIdx.x;
        const float dot = red[0][r] + red[1][r] + red[2][r] + red[3][r];
        const float al  = 1.0f / (1.0f + __expf(-dot));
        alphas[r] = al;
        if (alpha_out != nullptr) alpha_out[row0 + r] = al;
    }
    __syncthreads();

#pragma unroll
    for (int t = 0; t < TPW; ++t) {
        const int n = n0 + t * 16 + l15;
#pragma unroll
        for (int i = 0; i < 8; ++i) {
            const int r = i + half * 8;
            Out[(long)(row0 + r) * N + n] = acc[t][i] * alphas[r];
        }
    }
}

// ---------------------------------------------------------------------------
// Host launcher
// ---------------------------------------------------------------------------
extern "C" void kernel_launch(void* const* d_in, const int* in_sizes, int n_in,
                              void* d_out, int out_size, void* d_ws, size_t ws_size,
                              hipStream_t stream) {
    // Inputs in setup_inputs() order
    const float* feat     = (const float*)d_in[0];   // N_SRC1 x IN_DIM
    const float* W1       = (const float*)d_in[1];   // IN_DIM x HID_DIM
    const float* b1       = (const float*)d_in[2];   // HID_DIM
    const float* attn1    = (const float*)d_in[3];   // HID_DIM x 1
    const float* W2       = (const float*)d_in[4];   // HID_DIM x OUT_DIM
    const float* b2       = (const float*)d_in[5];   // OUT_DIM
    const float* attn2    = (const float*)d_in[6];   // OUT_DIM x 1
    const int*   src_idx1 = (const int*)d_in[7];     // E1
    const int*   dst_idx1 = (const int*)d_in[8];     // E1
    const int*   src_idx2 = (const int*)d_in[9];     // E2
    const int*   dst_idx2 = (const int*)d_in[10];    // E2

    float* out   = (float*)d_out;                    // 10000*128 h, then 10000 alpha
    float* out_h = out;
    float* out_a = out + (long)N_DST2 * OUT_DIM;

    // Workspace layout (floats). Zero-regions kept contiguous.
    float* ws = (float*)d_ws;
    size_t o = 0;
    float* deg1_src = ws + o;  o += N_SRC1;                   // rsqrt(out_deg) L1
    float* deg1_dst = ws + o;  o += N_DST1;                   // rsqrt(in_deg)  L1
    float* agg1     = ws + o;  o += (size_t)N_DST1 * IN_DIM;  // edge-summed feats
    float* h1       = ws + o;  o += (size_t)N_DST1 * HID_DIM; // gated layer-1 out
    float* deg2_src = ws + o;  o += N_SRC2;
    float* deg2_dst = ws + o;  o += N_DST2;
    float* agg2     = ws + o;  o += (size_t)N_DST2 * HID_DIM;

    const int TB = 256;

    // ---- Layer 1 ------------------------------------------------------
    {   // zero deg1_src | deg1_dst | agg1 (contiguous)
        int nz = N_SRC1 + N_DST1 + N_DST1 * IN_DIM;
        zero_f32_kernel<<<(nz + TB - 1) / TB, TB, 0, stream>>>(deg1_src, nz);
    }
    degree_kernel<<<(E1 + TB - 1) / TB, TB, 0, stream>>>(
        src_idx1, dst_idx1, deg1_src, deg1_dst, E1);
    rnorm_kernel<<<(N_SRC1 + TB - 1) / TB, TB, 0, stream>>>(deg1_src, N_SRC1);
    rnorm_kernel<<<(N_DST1 + TB - 1) / TB, TB, 0, stream>>>(deg1_dst, N_DST1);

    // 8 waves per block -> 8 edges per block
    aggregate_kernel<IN_DIM><<<(E1 + 7) / 8, TB, 0, stream>>>(
        feat, deg1_src, src_idx1, dst_idx1, agg1, E1);

    // h1 = gate( (agg1 * rn_dst) @ W1 + b1 )   (110000x128 @ 128x256, fused)
    gemm_gate_wmma_kernel<IN_DIM, 4><<<N_DST1 / 16, 128, 0, stream>>>(
        agg1, deg1_dst, W1, b1, attn1, h1, nullptr, HID_DIM);

    // ---- Layer 2 ------------------------------------------------------
    {   // zero deg2_src | deg2_dst | agg2 (contiguous)
        int nz = N_SRC2 + N_DST2 + N_DST2 * HID_DIM;
        zero_f32_kernel<<<(nz + TB - 1) / TB, TB, 0, stream>>>(deg2_src, nz);
    }
    degree_kernel<<<(E2 + TB - 1) / TB, TB, 0, stream>>>(
        src_idx2, dst_idx2, deg2_src, deg2_dst, E2);
    rnorm_kernel<<<(N_SRC2 + TB - 1) / TB, TB, 0, stream>>>(deg2_src, N_SRC2);
    rnorm_kernel<<<(N_DST2 + TB - 1) / TB, TB, 0, stream>>>(deg2_dst, N_DST2);

    aggregate_kernel<HID_DIM><<<(E2 + 7) / 8, TB, 0, stream>>>(
        h1, deg2_src, src_idx2, dst_idx2, agg2, E2);

    // out_h = gate( (agg2 * rn_dst) @ W2 + b2 ); out_a = alpha  (fused)
    gemm_gate_wmma_kernel<HID_DIM, 2><<<N_DST2 / 16, 128, 0, stream>>>(
        agg2, deg2_dst, W2, b2, attn2, out_h, out_a, OUT_DIM);

    (void)in_sizes; (void)n_in; (void)out_size; (void)ws_size;
}